// EdgeSIMPLEBatched_2482491097708
// MI455X (gfx1250) — compile-verified
//
#include <hip/hip_runtime.h>
#include <stdint.h>

// ---------------------------------------------------------------------------
// EdgeSIMPLEBatched forward: top-2048 k-hot mask over symmetrized scores.
// scores: (16, 512, 512, 4) f32.  out: (16, 512, 512, 4) f32 (0/1 symmetric).
// Algorithm: 3-stage radix select (11+11+10 bits) on monotonic u32 keys of
// scores + scores^T (upper triangle), then k-hot write + deterministic tie fix.
// CDNA5: tile loads use global_load_async_to_lds_b128 (ASYNCcnt) for the
// transpose staging; wave32 throughout.
// ---------------------------------------------------------------------------

#define USE_ASYNC 1

#define NM    512
#define ENS   4
#define BSZ   16
#define TS    130816u            // 512*511/2
#define KSEL  2048u
#define TILE  32
#define NT    (NM / TILE)        // 16 tiles per dim
#define NTP   (NT * (NT + 1) / 2) // 136 upper-tri tile pairs per batch
#define ROWS  (BSZ * ENS)        // 64
#define TIE_CAP 1024

// ---- workspace layout (bytes); total ~35.1 MB ----
#define OFF_KEYS  0ull
#define KEYS_BYTES ((unsigned long long)ROWS * TS * 4ull)        // 33,488,896
#define OFF_H1    (OFF_KEYS + KEYS_BYTES)                        // 64*2048 u32
#define OFF_H2    (OFF_H1 + (unsigned long long)ROWS * 2048 * 4)
#define OFF_H3    (OFF_H2 + (unsigned long long)ROWS * 2048 * 4) // 64*1024 u32
#define OFF_SEL   (OFF_H3 + (unsigned long long)ROWS * 1024 * 4) // 64*8 u32
#define OFF_TIEC  (OFF_SEL + (unsigned long long)ROWS * 8 * 4)   // 64 u32
#define OFF_TIEI  (OFF_TIEC + (unsigned long long)ROWS * 4)      // 64*TIE_CAP u32
// zeroed span: H1..TIEC inclusive
#define ZERO_U32  (ROWS*2048*2 + ROWS*1024 + ROWS*8 + ROWS)      // 328,256

__device__ __forceinline__ uint32_t fkey(float f) {
    uint32_t u = __float_as_uint(f);
    return (u & 0x80000000u) ? ~u : (u | 0x80000000u);  // monotonic: bigger float -> bigger key
}

__device__ __forceinline__ void ld_tile16(void* ldsPtr, const void* gPtr) {
#if USE_ASYNC
    // CDNA5 async global->LDS DMA, 16B per lane, tracked by ASYNCcnt.
    uint32_t l = (uint32_t)(uintptr_t)ldsPtr;   // low 32 bits of generic ptr == LDS byte offset
    uint64_t g = (uint64_t)(uintptr_t)gPtr;
    asm volatile("global_load_async_to_lds_b128 %0, %1, off"
                 :: "v"(l), "v"(g) : "memory");
#else
    *(float4*)ldsPtr = *(const float4*)gPtr;
#endif
}

__device__ __forceinline__ void wait_async_all() {
#if USE_ASYNC
    asm volatile("s_wait_asynccnt 0x0" ::: "memory");
#endif
}

__device__ __forceinline__ void decode_tilepair(int tp, int& TI, int& TJ) {
    int ti = 0, rem = tp;
    while (rem >= NT - ti) { rem -= NT - ti; ++ti; }
    TI = ti; TJ = ti + rem;                 // TI <= TJ
}

__device__ __forceinline__ uint32_t lin_idx(int i, int j) {
    // flat index of (i,j), i<j, in np.triu_indices(512, k=1) order
    return (uint32_t)i * (1023u - (uint32_t)i) / 2u + (uint32_t)(j - i - 1);
}

// --------------------------------------------------------------------------
__global__ __launch_bounds__(256)
void k_zero(uint32_t* p, int n) {
    for (int i = blockIdx.x * 256 + threadIdx.x; i < n; i += gridDim.x * 256) p[i] = 0u;
}

// --------------------------------------------------------------------------
// Pass 1: symmetrize via LDS-staged tile pairs, emit keys, top-11-bit histogram.
__global__ __launch_bounds__(256)
void k_sym_hist(const float* __restrict__ S, uint32_t* __restrict__ keys,
                uint32_t* __restrict__ h1) {
    __shared__ __align__(16) float tA[TILE][TILE + 1][ENS];
    __shared__ __align__(16) float tB[TILE][TILE + 1][ENS];
    __shared__ unsigned int hist[ENS * 2048];

    const int t = threadIdx.x;
    for (int i = t; i < ENS * 2048; i += 256) hist[i] = 0u;

    const int b = blockIdx.x / NTP;
    int TI, TJ; decode_tilepair(blockIdx.x % NTP, TI, TJ);
    const bool diag = (TI == TJ);

    const float* baseA = S + (((size_t)b * NM + TI * TILE) * NM + TJ * TILE) * ENS;
    const float* baseB = S + (((size_t)b * NM + TJ * TILE) * NM + TI * TILE) * ENS;

    // 32 rows x 32 chunks(16B) per tile; 256 threads -> 4 iterations
    #pragma unroll
    for (int it = 0; it < 4; ++it) {
        int lin = it * 256 + t;          // 0..1023
        int r = lin >> 5, c = lin & 31;
        ld_tile16(&tA[r][c][0], baseA + ((size_t)r * NM + c) * ENS);
        if (!diag) ld_tile16(&tB[r][c][0], baseB + ((size_t)r * NM + c) * ENS);
    }
    wait_async_all();
    __syncthreads();

    #pragma unroll
    for (int it = 0; it < 4; ++it) {
        int combo = it * 256 + t;        // 0..1023
        int x = combo >> 5, y = combo & 31;
        int i = TI * TILE + x, j = TJ * TILE + y;
        if (i < j) {
            uint32_t lin = lin_idx(i, j);
            #pragma unroll
            for (int e = 0; e < ENS; ++e) {
                float v = diag ? (tA[x][y][e] + tA[y][x][e])
                               : (tA[x][y][e] + tB[y][x][e]);
                uint32_t kk = fkey(v);
                atomicAdd(&hist[e * 2048 + (kk >> 21)], 1u);
                keys[(size_t)(b * ENS + e) * TS + lin] = kk;
            }
        }
    }
    __syncthreads();

    for (int idx = t; idx < ENS * 2048; idx += 256) {
        unsigned int v = hist[idx];
        if (v) atomicAdd(&h1[(size_t)(b * ENS + (idx >> 11)) * 2048 + (idx & 2047)], v);
    }
}

// --------------------------------------------------------------------------
// Scan a per-row histogram from the top; chain prefixes to exact threshold.
// Parallelized: LDS-stage the row, per-thread chunk sums, then one thread
// walks 256 chunk sums + <=8 bins (all LDS) instead of 2048 global loads.
// sel layout per row (8 u32): [0]=prefix1 [1]=need1 [2]=prefix2 [3]=need2
//                             [4]=threshold key T [5]=needTies
__global__ __launch_bounds__(256)
void k_scan(const uint32_t* __restrict__ hist, uint32_t* __restrict__ sel,
            int mode, int bins) {
    __shared__ uint32_t hloc[2048];
    __shared__ uint32_t csum[256];
    const int t = threadIdx.x;
    const int row = blockIdx.x;
    uint32_t* s = sel + row * 8;
    const uint32_t target = (mode == 0) ? KSEL : ((mode == 1) ? s[1] : s[3]);
    const uint32_t* h = hist + (size_t)row * bins;

    for (int i = t; i < bins; i += 256) hloc[i] = h[i];
    __syncthreads();

    const int per = bins >> 8;                // 8 (2048 bins) or 4 (1024 bins)
    uint32_t cs = 0;
    for (int k = 0; k < per; ++k) cs += hloc[t * per + k];
    csum[t] = cs;
    __syncthreads();

    if (t == 0) {
        uint32_t acc = 0; int c = 255;
        for (; c > 0; --c) {                  // find chunk from the top
            if (acc + csum[c] >= target) break;
            acc += csum[c];
        }
        int bin = c * per + per - 1;
        for (; bin > c * per; --bin) {        // find bin within chunk
            if (acc + hloc[bin] >= target) break;
            acc += hloc[bin];
        }
        if (mode == 0)      { s[0] = (uint32_t)bin; s[1] = target - acc; }
        else if (mode == 1) { s[2] = (s[0] << 11) | (uint32_t)bin; s[3] = target - acc; }
        else                { s[4] = (s[2] << 10) | (uint32_t)bin; s[5] = target - acc; }
    }
}

// --------------------------------------------------------------------------
// Refinement histogram over the linear key array (L2-resident).
__global__ __launch_bounds__(256)
void k_hist_refine(const uint32_t* __restrict__ keys, const uint32_t* __restrict__ sel,
                   uint32_t* __restrict__ hout, int mode) {
    __shared__ unsigned int hist[2048];
    const int bins = (mode == 1) ? 2048 : 1024;
    for (int i = threadIdx.x; i < bins; i += 256) hist[i] = 0u;
    __syncthreads();

    const int row = blockIdx.x >> 4;          // 16 chunks per row
    const int chunk = blockIdx.x & 15;
    const uint32_t pref = sel[row * 8 + ((mode == 1) ? 0 : 2)];
    const uint32_t* kr = keys + (size_t)row * TS;
    const uint32_t per = (TS + 15u) / 16u;
    const uint32_t s0 = chunk * per;
    const uint32_t s1 = (s0 + per < TS) ? (s0 + per) : TS;

    for (uint32_t idx = s0 + threadIdx.x; idx < s1; idx += 256) {
        uint32_t kk = kr[idx];
        if (mode == 1) { if ((kk >> 21) == pref) atomicAdd(&hist[(kk >> 10) & 2047u], 1u); }
        else           { if ((kk >> 10) == pref) atomicAdd(&hist[kk & 1023u], 1u); }
    }
    __syncthreads();
    uint32_t* ho = hout + (size_t)row * bins;
    for (int i = threadIdx.x; i < bins; i += 256) {
        unsigned int v = hist[i];
        if (v) atomicAdd(&ho[i], v);
    }
}

// --------------------------------------------------------------------------
// Output: k-hot (key > T), symmetric write via LDS tile + mirror, tie capture.
__global__ __launch_bounds__(256)
void k_out(const uint32_t* __restrict__ keys, const uint32_t* __restrict__ sel,
           uint32_t* __restrict__ tieCnt, uint32_t* __restrict__ tieIdx,
           float* __restrict__ out) {
    __shared__ __align__(16) float vt[TILE][TILE + 1][ENS];

    const int t = threadIdx.x;
    const int b = blockIdx.x / NTP;
    int TI, TJ; decode_tilepair(blockIdx.x % NTP, TI, TJ);
    const bool diag = (TI == TJ);

    uint32_t Tth[ENS];
    #pragma unroll
    for (int e = 0; e < ENS; ++e) Tth[e] = sel[(b * ENS + e) * 8 + 4];

    if (diag) {                                // unwritten (diag) slots must be 0
        for (int i = t; i < TILE * (TILE + 1) * ENS; i += 256)
            (&vt[0][0][0])[i] = 0.0f;
        __syncthreads();
    }

    #pragma unroll
    for (int it = 0; it < 4; ++it) {
        int combo = it * 256 + t;
        int x = combo >> 5, y = combo & 31;
        int i = TI * TILE + x, j = TJ * TILE + y;
        if (i < j) {
            uint32_t lin = lin_idx(i, j);
            #pragma unroll
            for (int e = 0; e < ENS; ++e) {
                uint32_t kk = keys[(size_t)(b * ENS + e) * TS + lin];
                float v = (kk > Tth[e]) ? 1.0f : 0.0f;
                if (kk == Tth[e]) {
                    int row = b * ENS + e;
                    uint32_t slot = atomicAdd(&tieCnt[row], 1u);
                    if (slot < TIE_CAP)
                        tieIdx[(size_t)row * TIE_CAP + slot] =
                            ((uint32_t)i << 16) | (uint32_t)j;
                }
                vt[x][y][e] = v;
                if (diag) vt[y][x][e] = v;     // symmetrize so dup stores agree
            }
        }
    }
    __syncthreads();

    // direct block (I,J): fully coalesced float4 stores
    #pragma unroll
    for (int it = 0; it < 4; ++it) {
        int combo = it * 256 + t;
        int x = combo >> 5, y = combo & 31;
        int i = TI * TILE + x, j = TJ * TILE + y;
        *(float4*)&out[(((size_t)b * NM + i) * NM + j) * ENS] = *(float4*)&vt[x][y][0];
    }
    // mirror block (J,I)
    if (!diag) {
        #pragma unroll
        for (int it = 0; it < 4; ++it) {
            int combo = it * 256 + t;
            int r = combo >> 5, c = combo & 31;
            int jj = TJ * TILE + r, ii = TI * TILE + c;
            *(float4*)&out[(((size_t)b * NM + jj) * NM + ii) * ENS] = *(float4*)&vt[c][r][0];
        }
    }
}

// --------------------------------------------------------------------------
// Deterministic tie resolution: sort captured indices, take lowest `needTies`.
__global__ void k_ties(const uint32_t* __restrict__ sel, const uint32_t* __restrict__ tieCnt,
                       uint32_t* __restrict__ tieIdx, float* __restrict__ out) {
    int row = blockIdx.x;
    uint32_t need = sel[row * 8 + 5];
    if (need == 0) return;
    uint32_t n = tieCnt[row]; if (n > TIE_CAP) n = TIE_CAP;
    uint32_t* L = tieIdx + (size_t)row * TIE_CAP;
    for (uint32_t a = 1; a < n; ++a) {         // insertion sort (n is tiny)
        uint32_t v = L[a]; int p = (int)a - 1;
        while (p >= 0 && L[p] > v) { L[p + 1] = L[p]; --p; }
        L[p + 1] = v;
    }
    if (need > n) need = n;
    int b = row >> 2, e = row & 3;
    for (uint32_t a = 0; a < need; ++a) {
        uint32_t p = L[a]; int i = (int)(p >> 16), j = (int)(p & 0xFFFFu);
        out[(((size_t)b * NM + i) * NM + j) * ENS + e] = 1.0f;
        out[(((size_t)b * NM + j) * NM + i) * ENS + e] = 1.0f;
    }
}

// --------------------------------------------------------------------------
extern "C" void kernel_launch(void* const* d_in, const int* in_sizes, int n_in,
                              void* d_out, int out_size, void* d_ws, size_t ws_size,
                              hipStream_t stream) {
    (void)in_sizes; (void)n_in; (void)out_size; (void)ws_size; // needs ~36 MB ws
    const float* S = (const float*)d_in[0];
    float* out = (float*)d_out;
    char* ws = (char*)d_ws;

    uint32_t* keys   = (uint32_t*)(ws + OFF_KEYS);
    uint32_t* h1     = (uint32_t*)(ws + OFF_H1);
    uint32_t* h2     = (uint32_t*)(ws + OFF_H2);
    uint32_t* h3     = (uint32_t*)(ws + OFF_H3);
    uint32_t* sel    = (uint32_t*)(ws + OFF_SEL);
    uint32_t* tieCnt = (uint32_t*)(ws + OFF_TIEC);
    uint32_t* tieIdx = (uint32_t*)(ws + OFF_TIEI);

    k_zero<<<256, 256, 0, stream>>>(h1, (int)ZERO_U32);
    k_sym_hist<<<BSZ * NTP, 256, 0, stream>>>(S, keys, h1);
    k_scan<<<ROWS, 256, 0, stream>>>(h1, sel, 0, 2048);
    k_hist_refine<<<ROWS * 16, 256, 0, stream>>>(keys, sel, h2, 1);
    k_scan<<<ROWS, 256, 0, stream>>>(h2, sel, 1, 2048);
    k_hist_refine<<<ROWS * 16, 256, 0, stream>>>(keys, sel, h3, 2);
    k_scan<<<ROWS, 256, 0, stream>>>(h3, sel, 2, 1024);
    k_out<<<BSZ * NTP, 256, 0, stream>>>(keys, sel, tieCnt, tieIdx, out);
    k_ties<<<ROWS, 1, 0, stream>>>(sel, tieCnt, tieIdx, out);
}